// EncoderLayerWithMoE_59141699666458
// MI455X (gfx1250) — compile-verified
//
#include <hip/hip_runtime.h>
#include <math.h>
#include <stdint.h>

#define BB   2
#define SS   2048
#define DD   768
#define HH   12
#define DHD  64
#define DFF_ 3072
#define EE   8
#define NTOK (BB*SS)   // 4096

typedef float v2f __attribute__((ext_vector_type(2)));
typedef float v8f __attribute__((ext_vector_type(8)));
typedef int   v4i __attribute__((ext_vector_type(4)));

__device__ __forceinline__ v8f wmma_f32(v2f a, v2f b, v8f c) {
  // D = A(16x4 f32) * B(4x16 f32) + C(16x16 f32)
  return __builtin_amdgcn_wmma_f32_16x16x4_f32(false, a, false, b, (short)0, c,
                                               false, false);
}

__device__ __forceinline__ v8f vzero8() {
  v8f z;
#pragma unroll
  for (int i = 0; i < 8; ++i) z[i] = 0.0f;
  return z;
}

// ---- CDNA5 async global->LDS staging (ASYNCcnt path), with sync fallback ----
#if __has_builtin(__builtin_amdgcn_global_load_async_to_lds_b128)
#define HAVE_ASYNC 1
#else
#define HAVE_ASYNC 0
#endif

#if HAVE_ASYNC
#if __has_builtin(__builtin_amdgcn_s_wait_asynccnt)
#define WAIT_ASYNC(n) __builtin_amdgcn_s_wait_asynccnt(n)
#else
#define WAIT_ASYNC(n) asm volatile("s_wait_asynccnt " #n ::: "memory")
#endif
typedef __attribute__((address_space(1))) v4i* gv4i_p;
typedef __attribute__((address_space(3))) v4i* lv4i_p;
__device__ __forceinline__ void stage_b128(const float* g, float* l) {
  // 16B global -> 16B LDS, tracked by ASYNCcnt
  __builtin_amdgcn_global_load_async_to_lds_b128(
      (gv4i_p)(uintptr_t)g,
      (lv4i_p)(uint32_t)(uintptr_t)l,
      0, 0);
}
#else
#define WAIT_ASYNC(n) do {} while (0)
__device__ __forceinline__ void stage_b128(const float* g, float* l) {
  *(float4*)l = *(const float4*)g;
}
#endif

// ---------------------------------------------------------------------------
// Generic LDS-tiled fp32 WMMA GEMM with async double-buffered staging.
// AMODE: 0 = direct rows, M fixed
//        1 = gathered rows via rowidx[], bounded by cnt[0]
//        2 = direct rows, bounded by cnt[0]
// OMODE: 0 = out[row*N+col] = v
//        1 = QKV head scatter: out[((b*H+h)*S+s)*DH+dh] = v
//        2 = MoE scatter-accum: out[tok*DD+col] += v * comb[tok*EE+expert]
// ---------------------------------------------------------------------------
template <int AMODE, int OMODE, bool RELU>
__launch_bounds__(128)
__global__ void gemm_wmma_k(const float* __restrict__ A,
                            const float* __restrict__ W,
                            const float* __restrict__ bias,
                            float* __restrict__ out,
                            int M, int Kd, int N,
                            const int* __restrict__ rowidx,
                            const int* __restrict__ cnt,
                            const float* __restrict__ comb, int expert) {
  // A rows padded to 20 floats: 80B stride -> 16B aligned + conflict-free
  __shared__ __align__(16) float As[2][64][20];
  __shared__ __align__(16) float Bs[2][16][64];
  const int tid  = threadIdx.x;
  const int lane = tid & 31, lid = lane & 15, half = lane >> 4;
  const int wave = tid >> 5;
  const int wm = (wave >> 1) * 32, wn = (wave & 1) * 32;
  const int m0 = blockIdx.y * 64, n0 = blockIdx.x * 64;

  int actM = M;
  if (AMODE != 0) actM = cnt[0];
  if (m0 >= actM) return;

  // precompute per-thread staging slots (2 A + 2 B float4 slots per thread).
  // out-of-range rows are clamped to row 0 (NOT predicated off) so every wave
  // issues exactly 4 async instructions per tile -> WAIT_ASYNC(4) is exact.
  int arow[2], ac4[2];
  long aoff[2];
#pragma unroll
  for (int s = 0; s < 2; ++s) {
    int slot = tid + s * 128;
    arow[s] = slot >> 2;
    ac4[s]  = (slot & 3) * 4;
    int gr = m0 + arow[s];
    int ar;
    if (AMODE == 1) ar = (gr < actM) ? rowidx[gr] : 0;
    else            ar = (gr < actM) ? gr : 0;
    aoff[s] = (long)ar * Kd + ac4[s];
  }
  int bkr[2], bc4[2];
#pragma unroll
  for (int s = 0; s < 2; ++s) {
    int slot = tid + s * 128;
    bkr[s] = slot >> 4;
    bc4[s] = (slot & 15) * 4;
  }

  auto issue_tile = [&](int t) {
    const int k0 = t * 16, bi = t & 1;
#pragma unroll
    for (int s = 0; s < 2; ++s)
      stage_b128(A + aoff[s] + k0, &As[bi][arow[s]][ac4[s]]);
#pragma unroll
    for (int s = 0; s < 2; ++s)
      stage_b128(W + (size_t)(k0 + bkr[s]) * N + n0 + bc4[s],
                 &Bs[bi][bkr[s]][bc4[s]]);
  };

  v8f acc[2][2];
#pragma unroll
  for (int i = 0; i < 2; ++i)
#pragma unroll
    for (int j = 0; j < 2; ++j) acc[i][j] = vzero8();

  const int nT = Kd / 16;
  issue_tile(0);
  for (int t = 0; t < nT; ++t) {
    if (t + 1 < nT) {
      issue_tile(t + 1);   // prefetch next tile into other buffer
      WAIT_ASYNC(4);       // current tile resident; next (4 per wave) in flight
    } else {
      WAIT_ASYNC(0);
    }
    __syncthreads();
    const int bi = t & 1;
#pragma unroll
    for (int kk = 0; kk < 16; kk += 4) {
      const int kb = kk + half * 2;  // lanes 0-15: K=kk,kk+1; 16-31: +2,+3
      v2f a[2], b[2];
#pragma unroll
      for (int i = 0; i < 2; ++i) {
        int row = wm + i * 16 + lid;
        a[i].x = As[bi][row][kb]; a[i].y = As[bi][row][kb + 1];
      }
#pragma unroll
      for (int j = 0; j < 2; ++j) {
        int col = wn + j * 16 + lid;
        b[j].x = Bs[bi][kb][col]; b[j].y = Bs[bi][kb + 1][col];
      }
#pragma unroll
      for (int i = 0; i < 2; ++i)
#pragma unroll
        for (int j = 0; j < 2; ++j)
          acc[i][j] = wmma_f32(a[i], b[j], acc[i][j]);
    }
    __syncthreads();  // all reads done before next prefetch overwrites buffer
  }

  // epilogue: C layout => vgpr r, lanes0-15 row r, lanes16-31 row r+8
#pragma unroll
  for (int i = 0; i < 2; ++i)
#pragma unroll
    for (int j = 0; j < 2; ++j)
#pragma unroll
      for (int r = 0; r < 8; ++r) {
        int grow = m0 + wm + i * 16 + half * 8 + r;
        int gcol = n0 + wn + j * 16 + lid;
        if (AMODE != 0 && grow >= actM) continue;
        float v = acc[i][j][r] + bias[gcol];
        if (RELU) v = fmaxf(v, 0.0f);
        if constexpr (OMODE == 0) {
          out[(size_t)grow * N + gcol] = v;
        } else if constexpr (OMODE == 1) {
          int b_ = grow / SS, s_ = grow % SS;
          int h_ = gcol / DHD, d_ = gcol % DHD;
          out[(((size_t)(b_ * HH + h_)) * SS + s_) * DHD + d_] = v;
        } else {
          int tok = rowidx[grow];
          out[(size_t)tok * DD + gcol] += v * comb[(size_t)tok * EE + expert];
        }
      }
}

// ---------------------------------------------------------------------------
// Flash attention: Q,K,V in [B*H, S, DH]; 64 queries per block, 4 waves,
// each wave owns 16 query rows; online softmax; ctx written as [B,S,D].
// ---------------------------------------------------------------------------
__launch_bounds__(128)
__global__ void attn_k(const float* __restrict__ Q, const float* __restrict__ K,
                       const float* __restrict__ V, float* __restrict__ ctx) {
  __shared__ __align__(16) float Qs[64][68];  // 272B stride: aligned, pad
  __shared__ __align__(16) float Ks[16][68];
  __shared__ __align__(16) float Vs[16][64];
  __shared__ float Ps[4][16][17];

  const int tid  = threadIdx.x;
  const int lane = tid & 31, lid = lane & 15, half = lane >> 4;
  const int wave = tid >> 5;
  const int bh = blockIdx.y;
  const int b = bh / HH, h = bh % HH;
  const int q0 = blockIdx.x * 64;
  const float* Qp = Q + (size_t)bh * SS * DHD;
  const float* Kp = K + (size_t)bh * SS * DHD;
  const float* Vp = V + (size_t)bh * SS * DHD;

  // stage Q tile (unscaled; 1/sqrt(DH) applied to scores)
#pragma unroll
  for (int s = 0; s < 8; ++s) {
    int slot = tid + s * 128;
    int row = slot >> 4, c4 = (slot & 15) * 4;
    stage_b128(Qp + (size_t)(q0 + row) * DHD + c4, &Qs[row][c4]);
  }

  float mrow[8], lrow[8];
  v8f O[4];
#pragma unroll
  for (int r = 0; r < 8; ++r) { mrow[r] = -1e30f; lrow[r] = 0.0f; }
#pragma unroll
  for (int t = 0; t < 4; ++t) O[t] = vzero8();

  const int mq = wave * 16;

  for (int kt = 0; kt < SS / 16; ++kt) {
    __syncthreads();   // prior iteration done with Ks/Vs/Ps
    const int kr0 = kt * 16;
#pragma unroll
    for (int s = 0; s < 2; ++s) {
      int slot = tid + s * 128;
      int row = slot >> 4, c4 = (slot & 15) * 4;
      stage_b128(Kp + (size_t)(kr0 + row) * DHD + c4, &Ks[row][c4]);
      stage_b128(Vp + (size_t)(kr0 + row) * DHD + c4, &Vs[row][c4]);
    }
    WAIT_ASYNC(0);     // covers Q on first iteration as well
    __syncthreads();

    // S_tile = Q(16xDH) * K^T(DHx16)
    v8f sa = vzero8();
#pragma unroll
    for (int kk = 0; kk < DHD; kk += 4) {
      const int kb = kk + half * 2;
      v2f a, bf;
      a.x = Qs[mq + lid][kb];  a.y = Qs[mq + lid][kb + 1];
      bf.x = Ks[lid][kb];      bf.y = Ks[lid][kb + 1];
      sa = wmma_f32(a, bf, sa);
    }

    // online softmax update, write P to LDS for transpose
#pragma unroll
    for (int r = 0; r < 8; ++r) {
      float sc = sa[r] * 0.125f;   // 1/sqrt(64)
      float tmax = sc;
      tmax = fmaxf(tmax, __shfl_xor(tmax, 1));
      tmax = fmaxf(tmax, __shfl_xor(tmax, 2));
      tmax = fmaxf(tmax, __shfl_xor(tmax, 4));
      tmax = fmaxf(tmax, __shfl_xor(tmax, 8));
      float nm = fmaxf(mrow[r], tmax);
      float p = __expf(sc - nm);
      float ps = p;
      ps += __shfl_xor(ps, 1);
      ps += __shfl_xor(ps, 2);
      ps += __shfl_xor(ps, 4);
      ps += __shfl_xor(ps, 8);
      float scl = __expf(mrow[r] - nm);
      lrow[r] = lrow[r] * scl + ps;
      mrow[r] = nm;
      O[0][r] *= scl; O[1][r] *= scl; O[2][r] *= scl; O[3][r] *= scl;
      Ps[wave][half * 8 + r][lid] = p;
    }
    __syncthreads();

    // O += P(16x16) * V(16x64)
#pragma unroll
    for (int nt = 0; nt < 4; ++nt) {
#pragma unroll
      for (int kk = 0; kk < 16; kk += 4) {
        const int kb = kk + half * 2;
        v2f a, bf;
        a.x = Ps[wave][lid][kb];       a.y = Ps[wave][lid][kb + 1];
        bf.x = Vs[kb][nt * 16 + lid];  bf.y = Vs[kb + 1][nt * 16 + lid];
        O[nt] = wmma_f32(a, bf, O[nt]);
      }
    }
  }

  // normalize + scatter to [B,S,D]
#pragma unroll
  for (int nt = 0; nt < 4; ++nt)
#pragma unroll
    for (int r = 0; r < 8; ++r) {
      int row = q0 + mq + half * 8 + r;
      int col = h * DHD + nt * 16 + lid;
      ctx[((size_t)b * SS + row) * DD + col] = O[nt][r] / lrow[r];
    }
}

// ---------------------------------------------------------------------------
// out = LayerNorm(x + res) * g + b ; one block per token (768 = 3*256)
// ---------------------------------------------------------------------------
__launch_bounds__(256)
__global__ void add_ln_k(const float* __restrict__ x, const float* __restrict__ r,
                         const float* __restrict__ g, const float* __restrict__ bt,
                         float* __restrict__ out) {
  __shared__ float red[256];
  const int tid = threadIdx.x;
  const size_t base = (size_t)blockIdx.x * DD;
  float a0 = x[base + tid] + r[base + tid];
  float a1 = x[base + tid + 256] + r[base + tid + 256];
  float a2 = x[base + tid + 512] + r[base + tid + 512];
  red[tid] = a0 + a1 + a2;
  __syncthreads();
  for (int off = 128; off > 0; off >>= 1) {
    if (tid < off) red[tid] += red[tid + off];
    __syncthreads();
  }
  float mean = red[0] * (1.0f / DD);
  __syncthreads();
  float d0 = a0 - mean, d1 = a1 - mean, d2 = a2 - mean;
  red[tid] = d0 * d0 + d1 * d1 + d2 * d2;
  __syncthreads();
  for (int off = 128; off > 0; off >>= 1) {
    if (tid < off) red[tid] += red[tid + off];
    __syncthreads();
  }
  float rstd = rsqrtf(red[0] * (1.0f / DD) + 1e-5f);
  out[base + tid]       = d0 * rstd * g[tid]       + bt[tid];
  out[base + tid + 256] = d1 * rstd * g[tid + 256] + bt[tid + 256];
  out[base + tid + 512] = d2 * rstd * g[tid + 512] + bt[tid + 512];
}

// ---------------------------------------------------------------------------
// Router: softmax over E=8, top-2, renorm gates -> combine; build expert lists
// ---------------------------------------------------------------------------
__launch_bounds__(256)
__global__ void router_k(const float* __restrict__ x, const float* __restrict__ rw,
                         const float* __restrict__ rb, float* __restrict__ comb,
                         int* __restrict__ counts, int* __restrict__ lists) {
  const int t = blockIdx.x * 256 + threadIdx.x;
  float lg[EE];
#pragma unroll
  for (int e = 0; e < EE; ++e) lg[e] = rb[e];
  const float* xr = x + (size_t)t * DD;
  for (int d = 0; d < DD; ++d) {
    float xv = xr[d];
    const float* wr = rw + d * EE;
#pragma unroll
    for (int e = 0; e < EE; ++e) lg[e] += xv * wr[e];
  }
  float mx = lg[0];
#pragma unroll
  for (int e = 1; e < EE; ++e) mx = fmaxf(mx, lg[e]);
  float p[EE], s = 0.0f;
#pragma unroll
  for (int e = 0; e < EE; ++e) { p[e] = __expf(lg[e] - mx); s += p[e]; }
  float inv = 1.0f / s;
#pragma unroll
  for (int e = 0; e < EE; ++e) p[e] *= inv;
  int i0 = 0;
#pragma unroll
  for (int e = 1; e < EE; ++e) if (p[e] > p[i0]) i0 = e;
  int i1 = (i0 == 0) ? 1 : 0;
#pragma unroll
  for (int e = 0; e < EE; ++e) if (e != i0 && p[e] > p[i1]) i1 = e;
  float gs = p[i0] + p[i1];
#pragma unroll
  for (int e = 0; e < EE; ++e)
    comb[(size_t)t * EE + e] =
        (e == i0) ? p[i0] / gs : ((e == i1) ? p[i1] / gs : 0.0f);
  int pos = atomicAdd(&counts[i0], 1);
  lists[i0 * NTOK + pos] = t;
  pos = atomicAdd(&counts[i1], 1);
  lists[i1 * NTOK + pos] = t;
}

// ---------------------------------------------------------------------------
// Aux loss: per-block fixed-order reduction of router2 probs + top-2 counts
// ---------------------------------------------------------------------------
__launch_bounds__(256)
__global__ void aux_partial_k(const float* __restrict__ x, const float* __restrict__ rw,
                              const float* __restrict__ rb,
                              float* __restrict__ pp, float* __restrict__ pc) {
  __shared__ float sp[256 * EE];
  __shared__ float sc[256 * EE];
  const int tid = threadIdx.x;
  const int t = blockIdx.x * 256 + tid;
  float lg[EE];
#pragma unroll
  for (int e = 0; e < EE; ++e) lg[e] = rb[e];
  const float* xr = x + (size_t)t * DD;
  for (int d = 0; d < DD; ++d) {
    float xv = xr[d];
    const float* wr = rw + d * EE;
#pragma unroll
    for (int e = 0; e < EE; ++e) lg[e] += xv * wr[e];
  }
  float mx = lg[0];
#pragma unroll
  for (int e = 1; e < EE; ++e) mx = fmaxf(mx, lg[e]);
  float p[EE], s = 0.0f;
#pragma unroll
  for (int e = 0; e < EE; ++e) { p[e] = __expf(lg[e] - mx); s += p[e]; }
  float inv = 1.0f / s;
#pragma unroll
  for (int e = 0; e < EE; ++e) p[e] *= inv;
  int i0 = 0;
#pragma unroll
  for (int e = 1; e < EE; ++e) if (p[e] > p[i0]) i0 = e;
  int i1 = (i0 == 0) ? 1 : 0;
#pragma unroll
  for (int e = 0; e < EE; ++e) if (e != i0 && p[e] > p[i1]) i1 = e;
#pragma unroll
  for (int e = 0; e < EE; ++e) {
    sp[tid * EE + e] = p[e];
    sc[tid * EE + e] = (e == i0 || e == i1) ? 1.0f : 0.0f;
  }
  __syncthreads();
  for (int off = 128; off > 0; off >>= 1) {
    if (tid < off) {
#pragma unroll
      for (int e = 0; e < EE; ++e) {
        sp[tid * EE + e] += sp[(tid + off) * EE + e];
        sc[tid * EE + e] += sc[(tid + off) * EE + e];
      }
    }
    __syncthreads();
  }
  if (tid == 0) {
#pragma unroll
    for (int e = 0; e < EE; ++e) {
      pp[blockIdx.x * EE + e] = sp[e];
      pc[blockIdx.x * EE + e] = sc[e];
    }
  }
}

__global__ void aux_final_k(const float* __restrict__ pp,
                            const float* __restrict__ pc,
                            float* __restrict__ out) {
  if (threadIdx.x == 0 && blockIdx.x == 0) {
    float aux = 0.0f;
    for (int e = 0; e < EE; ++e) {
      float ps = 0.0f, cs = 0.0f;
      for (int b = 0; b < NTOK / 256; ++b) {
        ps += pp[b * EE + e];
        cs += pc[b * EE + e];
      }
      float frac = cs / (float)(NTOK * 2);
      float prob = ps / (float)NTOK;
      aux += frac * prob;
    }
    out[0] = (float)EE * aux;
  }
}

// ---------------------------------------------------------------------------
extern "C" void kernel_launch(void* const* d_in, const int* in_sizes, int n_in,
                              void* d_out, int out_size, void* d_ws, size_t ws_size,
                              hipStream_t stream) {
  const float* x   = (const float*)d_in[0];
  const float* wq  = (const float*)d_in[1];
  const float* bq  = (const float*)d_in[2];
  const float* wk  = (const float*)d_in[3];
  const float* bk  = (const float*)d_in[4];
  const float* wv  = (const float*)d_in[5];
  const float* bv  = (const float*)d_in[6];
  const float* wo  = (const float*)d_in[7];
  const float* bo  = (const float*)d_in[8];
  const float* n1g = (const float*)d_in[9];
  const float* n1b = (const float*)d_in[10];
  const float* rw  = (const float*)d_in[11];
  const float* rb  = (const float*)d_in[12];
  const float* ew1 = (const float*)d_in[13];
  const float* eb1 = (const float*)d_in[14];
  const float* ew2 = (const float*)d_in[15];
  const float* eb2 = (const float*)d_in[16];
  const float* n2g = (const float*)d_in[17];
  const float* n2b = (const float*)d_in[18];
  float* out = (float*)d_out;

  const size_t NQ = (size_t)NTOK * DD;   // 3,145,728 floats
  float* w    = (float*)d_ws;
  float* q    = w;
  float* kbf  = q + NQ;
  float* vbf  = kbf + NQ;
  float* ctx  = vbf + NQ;
  float* tmp  = ctx + NQ;
  float* x1   = tmp + NQ;
  float* moe  = x1 + NQ;
  float* hbuf = moe + NQ;                       // NTOK*DFF
  float* comb = hbuf + (size_t)NTOK * DFF_;     // NTOK*EE
  float* pp   = comb + (size_t)NTOK * EE;       // 16*EE
  float* pc   = pp + (NTOK / 256) * EE;
  int* counts = (int*)(pc + (NTOK / 256) * EE);
  int* lists  = counts + EE;                    // EE*NTOK ints

  dim3 blk128(128), blk256(256);
  dim3 gProj(DD / 64, NTOK / 64);   // (12, 64)

  // QKV projections -> [B,H,S,DH]
  gemm_wmma_k<0, 1, false><<<gProj, blk128, 0, stream>>>(
      x, wq, bq, q, NTOK, DD, DD, nullptr, nullptr, nullptr, 0);
  gemm_wmma_k<0, 1, false><<<gProj, blk128, 0, stream>>>(
      x, wk, bk, kbf, NTOK, DD, DD, nullptr, nullptr, nullptr, 0);
  gemm_wmma_k<0, 1, false><<<gProj, blk128, 0, stream>>>(
      x, wv, bv, vbf, NTOK, DD, DD, nullptr, nullptr, nullptr, 0);

  // flash attention -> ctx [B,S,D]
  attn_k<<<dim3(SS / 64, BB * HH), blk128, 0, stream>>>(q, kbf, vbf, ctx);

  // out projection + residual LN
  gemm_wmma_k<0, 0, false><<<gProj, blk128, 0, stream>>>(
      ctx, wo, bo, tmp, NTOK, DD, DD, nullptr, nullptr, nullptr, 0);
  add_ln_k<<<NTOK, blk256, 0, stream>>>(x, tmp, n1g, n1b, x1);

  // router + top-2 expert token lists
  (void)hipMemsetAsync(counts, 0, EE * sizeof(int), stream);
  (void)hipMemsetAsync(moe, 0, NQ * sizeof(float), stream);
  router_k<<<NTOK / 256, blk256, 0, stream>>>(x1, rw, rb, comb, counts, lists);

  // sparse MoE: only routed tokens per expert (gather GEMM + scatter GEMM)
  for (int e = 0; e < EE; ++e) {
    gemm_wmma_k<1, 0, true><<<dim3(DFF_ / 64, NTOK / 64), blk128, 0, stream>>>(
        x1, ew1 + (size_t)e * DD * DFF_, eb1 + (size_t)e * DFF_, hbuf,
        NTOK, DD, DFF_, lists + (size_t)e * NTOK, counts + e, nullptr, e);
    gemm_wmma_k<2, 2, false><<<dim3(DD / 64, NTOK / 64), blk128, 0, stream>>>(
        hbuf, ew2 + (size_t)e * DFF_ * DD, eb2 + (size_t)e * DD, moe,
        NTOK, DFF_, DD, lists + (size_t)e * NTOK, counts + e, comb, e);
  }

  // final LN -> x2 (d_out), then aux loss from second router pass
  add_ln_k<<<NTOK, blk256, 0, stream>>>(x1, moe, n2g, n2b, out);
  aux_partial_k<<<NTOK / 256, blk256, 0, stream>>>(out, rw, rb, pp, pc);
  aux_final_k<<<1, 32, 0, stream>>>(pp, pc, out + NQ);
}